// VGAE_8160437862467
// MI455X (gfx1250) — compile-verified
//
#include <hip/hip_runtime.h>
#include <math.h>

typedef __attribute__((ext_vector_type(2))) float v2f;
typedef __attribute__((ext_vector_type(8))) float v8f;

#define IN_CH 256
#define OUT_CH 64

// ---------------- degree / normalization ----------------

__global__ void init_deg_kernel(float* __restrict__ deg, int n) {
  int i = blockIdx.x * blockDim.x + threadIdx.x;
  if (i < n) deg[i] = 1.0f;  // self-loop contributes 1 to in-degree
}

__global__ void count_deg_kernel(const int* __restrict__ dst,
                                 float* __restrict__ deg, int e) {
  int i = blockIdx.x * blockDim.x + threadIdx.x;
  if (i < e) atomicAdd(&deg[dst[i]], 1.0f);  // exact: counts < 2^24
}

__global__ void dinv_kernel(float* __restrict__ deg, int n) {
  int i = blockIdx.x * blockDim.x + threadIdx.x;
  if (i < n) deg[i] = rsqrtf(deg[i]);  // deg >= 1 always
}

// ---------------- WMMA GEMM: h' = dinv[row] * (x @ W^T) ----------------
// Block = 256 threads = 8 waves. Block covers 32 rows x 64 cols.
// Wave w: row-tile w/4 (16 rows), col-tile w%4 (16 cols).
// V_WMMA_F32_16X16X4_F32: A is 16x4 f32 (2 VGPR), B is 4x16 f32 (2 VGPR),
// C/D 16x16 f32 (8 VGPR). A lane layout: lanes 0-15 hold K=0,1; lanes
// 16-31 hold K=2,3 (M = lane%16). B mirrored with N = lane%16.
__global__ __launch_bounds__(256) void gemm_wmma_kernel(
    const float* __restrict__ x, const float* __restrict__ Wmu,
    const float* __restrict__ Wlv, const float* __restrict__ dinv,
    float* __restrict__ hmu, float* __restrict__ hlv,
    float* __restrict__ accmu, float* __restrict__ acclv) {
  const int lane = threadIdx.x & 31;
  const int wave = threadIdx.x >> 5;        // 0..7
  const int row0 = blockIdx.x * 32 + (wave >> 2) * 16;
  const int col0 = (wave & 3) * 16;

  const int m     = lane & 15;              // row (A) / col (B) within tile
  const int khalf = (lane >> 4) * 2;        // 0 or 2

  const float* __restrict__ xrow  = x   + (size_t)(row0 + m) * IN_CH + khalf;
  const float* __restrict__ wmrow = Wmu + (size_t)(col0 + m) * IN_CH + khalf;
  const float* __restrict__ wlrow = Wlv + (size_t)(col0 + m) * IN_CH + khalf;

  v8f cmu = {};
  v8f clv = {};
#pragma unroll 4
  for (int k = 0; k < IN_CH; k += 4) {
    v2f a  = *(const v2f*)(xrow + k);
    v2f bm = *(const v2f*)(wmrow + k);
    v2f bl = *(const v2f*)(wlrow + k);
    cmu = __builtin_amdgcn_wmma_f32_16x16x4_f32(false, a, false, bm,
                                                (short)0, cmu, false, false);
    clv = __builtin_amdgcn_wmma_f32_16x16x4_f32(false, a, false, bl,
                                                (short)0, clv, false, false);
  }

  // D layout: VGPR v, lane L -> row = v + 8*(L/16), col = L%16
  const int colw  = lane & 15;
  const int rbase = (lane >> 4) * 8;
#pragma unroll
  for (int v = 0; v < 8; ++v) {
    const int row = row0 + rbase + v;
    const float s = dinv[row];
    const float hm = cmu[v] * s;
    const float hl = clv[v] * s;
    const size_t idx = (size_t)row * OUT_CH + col0 + colw;
    hmu[idx] = hm;
    hlv[idx] = hl;
    accmu[idx] = hm;  // acc starts at self-loop term h'[i]
    acclv[idx] = hl;
  }
}

// ---------------- edge scatter: acc[dst] += h'[src] ----------------
// 16 threads per edge, 4 channels each (float4 gather, resolves in L2).
__global__ __launch_bounds__(256) void scatter_kernel(
    const int* __restrict__ src, const int* __restrict__ dst,
    const float* __restrict__ hmu, const float* __restrict__ hlv,
    float* __restrict__ accmu, float* __restrict__ acclv, int e) {
  const unsigned t = blockIdx.x * blockDim.x + threadIdx.x;
  const int ei = (int)(t >> 4);
  if (ei >= e) return;
  const int cg = (int)(t & 15) * 4;
  const int s = src[ei];
  const int d = dst[ei];

  const float4 hm = *(const float4*)(hmu + (size_t)s * OUT_CH + cg);
  const float4 hl = *(const float4*)(hlv + (size_t)s * OUT_CH + cg);
  float* pm = accmu + (size_t)d * OUT_CH + cg;
  float* pl = acclv + (size_t)d * OUT_CH + cg;
  atomicAdd(pm + 0, hm.x);
  atomicAdd(pm + 1, hm.y);
  atomicAdd(pm + 2, hm.z);
  atomicAdd(pm + 3, hm.w);
  atomicAdd(pl + 0, hl.x);
  atomicAdd(pl + 1, hl.y);
  atomicAdd(pl + 2, hl.z);
  atomicAdd(pl + 3, hl.w);
}

// ---------------- finalize: scale, bias, relu, reparameterize ----------------
__global__ void finalize_kernel(const float* __restrict__ accmu,
                                const float* __restrict__ acclv,
                                const float* __restrict__ dinv,
                                const float* __restrict__ bmu,
                                const float* __restrict__ blv,
                                const float* __restrict__ eps,
                                float* __restrict__ out, int n) {
  const size_t total = (size_t)n * OUT_CH;
  const size_t i = (size_t)blockIdx.x * blockDim.x + threadIdx.x;
  if (i >= total) return;
  const int node = (int)(i >> 6);
  const int c = (int)(i & 63);
  const float s = dinv[node];
  const float mu = fmaxf(fmaf(accmu[i], s, bmu[c]), 0.0f);
  const float lv = fmaxf(fmaf(acclv[i], s, blv[c]), 0.0f);
  const float z = fmaf(eps[i], expf(0.5f * lv), mu);
  out[i] = z;                 // z
  out[i + total] = mu;        // mu
  out[i + 2 * total] = lv;    // logvar
}

// ---------------- host-side launch ----------------

extern "C" void kernel_launch(void* const* d_in, const int* in_sizes, int n_in,
                              void* d_out, int out_size, void* d_ws,
                              size_t ws_size, hipStream_t stream) {
  const float* x   = (const float*)d_in[0];
  const int* edge  = (const int*)d_in[1];   // int32 indices (JAX default x64 off)
  const float* Wmu = (const float*)d_in[2];
  const float* bmu = (const float*)d_in[3];
  const float* Wlv = (const float*)d_in[4];
  const float* blv = (const float*)d_in[5];
  const float* eps = (const float*)d_in[6];
  float* out = (float*)d_out;

  const int N = in_sizes[0] / IN_CH;  // 100000 (divisible by 32)
  const int E = in_sizes[1] / 2;      // 3200000
  const int* srcI = edge;
  const int* dstI = edge + E;

  // workspace carve-out
  char* ws = (char*)d_ws;
  size_t ofs = 0;
  auto carve = [&](size_t bytes) -> void* {
    void* p = ws + ofs;
    ofs = (ofs + bytes + 255) & ~(size_t)255;
    return p;
  };
  float* dinv  = (float*)carve((size_t)N * sizeof(float));
  float* hmu   = (float*)carve((size_t)N * OUT_CH * sizeof(float));
  float* hlv   = (float*)carve((size_t)N * OUT_CH * sizeof(float));
  float* accmu = (float*)carve((size_t)N * OUT_CH * sizeof(float));
  float* acclv = (float*)carve((size_t)N * OUT_CH * sizeof(float));

  const int TB = 256;

  init_deg_kernel<<<(N + TB - 1) / TB, TB, 0, stream>>>(dinv, N);
  count_deg_kernel<<<(E + TB - 1) / TB, TB, 0, stream>>>(dstI, dinv, E);
  dinv_kernel<<<(N + TB - 1) / TB, TB, 0, stream>>>(dinv, N);

  gemm_wmma_kernel<<<N / 32, TB, 0, stream>>>(x, Wmu, Wlv, dinv, hmu, hlv,
                                              accmu, acclv);

  const size_t scat_threads = (size_t)E * 16;
  scatter_kernel<<<(unsigned)((scat_threads + TB - 1) / TB), TB, 0, stream>>>(
      srcI, dstI, hmu, hlv, accmu, acclv, E);

  const size_t fin_threads = (size_t)N * OUT_CH;
  finalize_kernel<<<(unsigned)((fin_threads + TB - 1) / TB), TB, 0, stream>>>(
      accmu, acclv, dinv, bmu, blv, eps, out, N);
}